// TensorProduct_71648644432354
// MI455X (gfx1250) — compile-verified
//
#include <hip/hip_runtime.h>

typedef __attribute__((ext_vector_type(2))) float v2f;
typedef __attribute__((ext_vector_type(8))) float v8f;

// LDS tile: 16 transform-lines (r) x 1024 elements (e), column-major [e][r]
// with per-row XOR swizzle; DWORD index = (e<<4) + ((r^e)&15).
// Since the swizzle only sees e&15, all compute offsets are loop-invariant.
#define LDS_AT(t, e, r) (t)[(((e) << 4)) + ((((r) ^ (e))) & 15)]

__device__ __forceinline__ unsigned lds_off(const void* p) {
    // Generic pointers to LDS carry the LDS byte offset in the low 32 bits.
    return (unsigned)(uintptr_t)p;
}

// ---------------------------------------------------------------------------
// Compose levels 0..3 of one twiddle set into 64 dense 16x16 block matrices.
// Amat[(k*16 + i)*16 + j] = coefficient of input j in output i of block k.
// ---------------------------------------------------------------------------
__global__ void compose16(const float* __restrict__ tw, float* __restrict__ Am) {
    int tid = blockIdx.x * blockDim.x + threadIdx.x;   // 0..1023
    int k = tid >> 4, j = tid & 15;
    float v[16];
#pragma unroll
    for (int p = 0; p < 16; ++p) v[p] = (p == j) ? 1.0f : 0.0f;
#pragma unroll
    for (int l = 0; l < 4; ++l) {
        int s = 1 << l;
#pragma unroll
        for (int p = 0; p < 16; ++p) {
            if (((p >> l) & 1) == 0) {
                int i0 = p, i1 = p + s;
                int g0 = (k << 4) + i0;
                int twIdx = ((g0 >> (l + 1)) << l) + (g0 & (s - 1));
                const float* tm = tw + (((l * 512) + twIdx) << 2);
                float x0 = v[i0], x1 = v[i1];
                v[i0] = tm[0] * x0 + tm[1] * x1;
                v[i1] = tm[2] * x0 + tm[3] * x1;
            }
        }
    }
    for (int i = 0; i < 16; ++i) Am[((k * 16 + i) << 4) + j] = v[i];
}

// ---------------------------------------------------------------------------
// Compose levels 4..9 into 16 dense 64x64 matrices (one per residue c = e%16).
// ---------------------------------------------------------------------------
__global__ void compose64(const float* __restrict__ tw, float* __restrict__ Bm) {
    int tid = blockIdx.x * blockDim.x + threadIdx.x;   // 0..1023
    int c = tid >> 6, j = tid & 63;
    float v[64];
    for (int p = 0; p < 64; ++p) v[p] = (p == j) ? 1.0f : 0.0f;
#pragma unroll
    for (int l = 4; l < 10; ++l) {
        int s = 1 << l, sr = s >> 4;
        for (int r0 = 0; r0 < 64; ++r0) {
            if (((r0 >> (l - 4)) & 1) == 0) {
                int r1 = r0 + sr;
                int g0 = (r0 << 4) + c;
                int twIdx = ((g0 >> (l + 1)) << l) + (g0 & (s - 1));
                const float* tm = tw + (((l * 512) + twIdx) << 2);
                float x0 = v[r0], x1 = v[r1];
                v[r0] = tm[0] * x0 + tm[1] * x1;
                v[r1] = tm[2] * x0 + tm[3] * x1;
            }
        }
    }
    for (int i = 0; i < 64; ++i) Bm[(c * 64 + i) * 64 + j] = v[i];
}

// ---------------------------------------------------------------------------
// In-LDS butterfly on a 16-line x 1024 tile via V_WMMA_F32_16X16X4_F32.
// All LDS/global addresses are (per-lane base + compile-time immediate).
// ---------------------------------------------------------------------------
__device__ __forceinline__ void tile_butterfly(float* tile,
                                               const float* __restrict__ Am,
                                               const float* __restrict__ Bm) {
    const int lane = threadIdx.x & 31;
    const int w    = threadIdx.x >> 5;
    const int n    = lane & 15;     // N (line) index; also M index for A-frags
    const int hi   = lane >> 4;     // half-wave select

    // ---- Stage A: levels 0-3 as 64 block-diagonal 16x16 GEMMs -------------
    // Per-lane invariant swizzled offsets within one 256-DWORD block.
    int bfo[8];                     // B-frag: f = 4q + 2hi + t
#pragma unroll
    for (int q = 0; q < 4; ++q)
#pragma unroll
        for (int t = 0; t < 2; ++t) {
            int f = 4 * q + 2 * hi + t;
            bfo[2 * q + t] = (f << 4) + ((n ^ f) & 15);
        }
    int dfo[8];                     // D-store: f = vv + 8hi
#pragma unroll
    for (int vv = 0; vv < 8; ++vv) {
        int f = vv + 8 * hi;
        dfo[vv] = (f << 4) + ((n ^ f) & 15);
    }
    float* tA = tile + (w << 12);                       // wave's 16 blocks
    const float* amB = Am + (w << 12) + (n << 4) + 2 * hi;
#pragma unroll
    for (int kb2 = 0; kb2 < 16; ++kb2) {                // block within wave
        const int bo = kb2 << 8;                        // DWORDs (imm offset)
        v2f bf[4];
#pragma unroll
        for (int q = 0; q < 4; ++q) {
            bf[q].x = tA[bo + bfo[2 * q]];
            bf[q].y = tA[bo + bfo[2 * q + 1]];
        }
        v8f acc = {0.f, 0.f, 0.f, 0.f, 0.f, 0.f, 0.f, 0.f};
#pragma unroll
        for (int q = 0; q < 4; ++q) {
            v2f af = *(const v2f*)(amB + bo + 4 * q);
            acc = __builtin_amdgcn_wmma_f32_16x16x4_f32(
                false, af, false, bf[q], (short)0, acc, false, false);
        }
#pragma unroll
        for (int vv = 0; vv < 8; ++vv)
            tA[bo + dfo[vv]] = acc[vv];
    }
    __syncthreads();

    // ---- Stage B: levels 4-9 as 16 x (64x64 . 64x16) GEMMs ----------------
    const float* bmN = Bm + (n << 6) + 2 * hi;
#pragma unroll
    for (int cb2 = 0; cb2 < 4; ++cb2) {
        const int cb = w * 4 + cb2;
        float* tc  = tile + (cb << 4) + ((n ^ cb) & 15);  // column base
        float* tcB = tc + (hi << 9);                      // + e-part of 2hi
        float* tcD = tc + (hi << 11);                     // + e-part of 8hi
        v2f bf[16];                                       // full K=64 resident
#pragma unroll
        for (int ch = 0; ch < 16; ++ch) {                 // e = 16*(4ch+t+2hi)+cb
            bf[ch].x = tcB[(ch << 10)];
            bf[ch].y = tcB[(ch << 10) + 256];
        }
        const float* bmc = bmN + (cb << 12);
#pragma unroll
        for (int mt = 0; mt < 4; ++mt) {
            v8f acc = {0.f, 0.f, 0.f, 0.f, 0.f, 0.f, 0.f, 0.f};
#pragma unroll
            for (int ch = 0; ch < 16; ++ch) {
                v2f af = *(const v2f*)(bmc + (mt << 10) + 4 * ch);
                acc = __builtin_amdgcn_wmma_f32_16x16x4_f32(
                    false, af, false, bf[ch], (short)0, acc, false, false);
            }
#pragma unroll
            for (int vv = 0; vv < 8; ++vv)                // e = 16*(mt*16+vv+8hi)+cb
                tcD[(mt << 12) + (vv << 8)] = acc[vv];
        }
    }
}

// ---------------------------------------------------------------------------
// Transform 1: butterfly along N1 (contiguous axis). One WG per 16 rows.
// Contiguous staging -> b128 global loads/stores + immediate-offset DS ops.
// ---------------------------------------------------------------------------
__global__ __launch_bounds__(128) void bfly_rows(const float* __restrict__ src,
                                                 float* __restrict__ dst,
                                                 const float* __restrict__ Am,
                                                 const float* __restrict__ Bm) {
    __shared__ float tile[1024 * 16];
    const int b    = blockIdx.x >> 6;
    const int row0 = (blockIdx.x & 63) << 4;
    const size_t base = (size_t)b * (1024u * 1024u) + (size_t)row0 * 1024u;
    const int r  = threadIdx.x >> 3;   // 0..15 line
    const int e8 = threadIdx.x & 7;    // float4 slot
    // e = 4*(e8 + 8it) + q: e&15 = (4e8+q)&15 -> per-q invariant swizzle.
    float* tS[4];
#pragma unroll
    for (int q = 0; q < 4; ++q)
        tS[q] = tile + (e8 << 6) + (q << 4) + ((r ^ (4 * e8 + q)) & 15);
    const float* gp = src + base + (size_t)r * 1024 + 4 * e8;
#pragma unroll
    for (int it = 0; it < 32; ++it) {
        float4 d = *(const float4*)(gp + 32 * it);
        tS[0][it << 9] = d.x;  tS[1][it << 9] = d.y;
        tS[2][it << 9] = d.z;  tS[3][it << 9] = d.w;
    }
    __syncthreads();
    tile_butterfly(tile, Am, Bm);
    __syncthreads();
    float* op = dst + base + (size_t)r * 1024 + 4 * e8;
#pragma unroll
    for (int it = 0; it < 32; ++it) {
        float4 d;
        d.x = tS[0][it << 9];  d.y = tS[1][it << 9];
        d.z = tS[2][it << 9];  d.w = tS[3][it << 9];
        *(float4*)(op + 32 * it) = d;
    }
}

// ---------------------------------------------------------------------------
// Transform 2: butterfly along N2 (stride-1024 axis), in place on d_out.
// Transposed gather staged with CDNA5 async global->LDS (ASYNCcnt-tracked),
// drained with s_wait_asynccnt before the workgroup barrier.
// ---------------------------------------------------------------------------
__global__ __launch_bounds__(128) void bfly_cols(float* __restrict__ data,
                                                 const float* __restrict__ Am,
                                                 const float* __restrict__ Bm) {
    __shared__ float tile[1024 * 16];
    const int b    = blockIdx.x >> 6;
    const int col0 = (blockIdx.x & 63) << 4;
    const size_t base = (size_t)b * (1024u * 1024u) + (size_t)col0;

    // Async staging-in: thread -> (line j, rows i = i8 + 8*it).
    {
        const int j  = threadIdx.x & 15;
        const int i8 = threadIdx.x >> 4;      // 0..7
        const float* gp = data + base + (size_t)i8 * 1024 + j;
        const unsigned lb = lds_off(tile);
#pragma unroll 8
        for (int it = 0; it < 128; ++it) {
            int i = i8 + 8 * it;
            unsigned l = lb + ((((unsigned)i << 4) + (((unsigned)(j ^ i)) & 15)) << 2);
            unsigned long long ga =
                (unsigned long long)(uintptr_t)(gp + (size_t)(8 * it) * 1024);
            asm volatile("global_load_async_to_lds_b32 %0, %1, off"
                         :: "v"(l), "v"(ga) : "memory");
        }
        asm volatile("s_wait_asynccnt 0x0" ::: "memory");
    }
    __syncthreads();
    tile_butterfly(tile, Am, Bm);
    __syncthreads();

    // Store-back: registers + b128 global stores (contiguous along N1 cols).
    const int j4 = threadIdx.x & 3;
    const int i0 = threadIdx.x >> 2;   // 0..31
    float* tS[4];
#pragma unroll
    for (int q = 0; q < 4; ++q)        // e = i0 + 1024it: e&15 = i0&15
        tS[q] = tile + (i0 << 4) + (((4 * j4 + q) ^ i0) & 15);
    float* op = data + base + (size_t)i0 * 1024 + 4 * j4;
#pragma unroll
    for (int it = 0; it < 32; ++it) {
        float4 d;
        d.x = tS[0][it << 14];  d.y = tS[1][it << 14];
        d.z = tS[2][it << 14];  d.w = tS[3][it << 14];
        *(float4*)(op + (size_t)(32 * it) * 1024) = d;
    }
}

extern "C" void kernel_launch(void* const* d_in, const int* in_sizes, int n_in,
                              void* d_out, int out_size, void* d_ws, size_t ws_size,
                              hipStream_t stream) {
    const float* x   = (const float*)d_in[0];   // (32, 1024, 1024) fp32
    const float* tw1 = (const float*)d_in[1];   // (10, 512, 2, 2)
    const float* tw2 = (const float*)d_in[2];   // (10, 512, 2, 2)
    float* out = (float*)d_out;

    // Workspace: composed butterfly matrices (640 KB total).
    float* A1 = (float*)d_ws;          // 64 * 16 * 16
    float* B1 = A1 + 64 * 256;         // 16 * 64 * 64
    float* A2 = B1 + 16 * 4096;        // 64 * 16 * 16
    float* B2 = A2 + 64 * 256;         // 16 * 64 * 64
    (void)in_sizes; (void)n_in; (void)out_size; (void)ws_size;

    compose16<<<4, 256, 0, stream>>>(tw1, A1);
    compose64<<<4, 256, 0, stream>>>(tw1, B1);
    compose16<<<4, 256, 0, stream>>>(tw2, A2);
    compose64<<<4, 256, 0, stream>>>(tw2, B2);

    bfly_rows<<<32 * 64, 128, 0, stream>>>(x, out, A1, B1);
    bfly_cols<<<32 * 64, 128, 0, stream>>>(out, A2, B2);
}